// RGAT_29145648070963
// MI455X (gfx1250) — compile-verified
//
#include <hip/hip_runtime.h>

#define N_NODES 100000
#define E_EDGES 250000
#define R_REL   4
#define H_HEADS 4
#define D_DIM   32
#define F_IN    256
#define HID     128
#define F_OUT   64
#define SLOPE   0.2f

typedef __attribute__((ext_vector_type(2))) float v2f;
typedef __attribute__((ext_vector_type(8))) float v8f;

// ---------------------------------------------------------------------------
// WMMA fp32 GEMM: C[M x ncols] = A[M x K] * B[K x ncols] (+ bias)
// One block handles a 16-row strip; 8 waves each own a 16-wide column tile.
// Uses V_WMMA_F32_16X16X4_F32 (wave32).  M, K, ncols all multiples of 16/4.
// ---------------------------------------------------------------------------
__global__ __launch_bounds__(256) void gemm_wmma_f32(
    const float* __restrict__ A, const float* __restrict__ B,
    const float* __restrict__ bias, float* __restrict__ C,
    int K, int ncols)
{
    __shared__ float As[16 * 260];          // K<=256, +4 pad keeps 16B row align
    const int lds = K + 4;
    const int row0 = blockIdx.x << 4;
    const int tid = threadIdx.x;

    // cooperative load of the 16 x K A-tile (float4 vectorized)
    const int total = 16 * K;
    for (int i = tid * 4; i < total; i += 256 * 4) {
        int r = i / K, c = i - r * K;
        float4 v = *(const float4*)(A + (size_t)(row0 + r) * K + c);
        *(float4*)(As + r * lds + c) = v;
    }
    __syncthreads();

    const int wave = tid >> 5;
    const int lane = tid & 31;
    const int ntiles = ncols >> 4;
    if (wave < ntiles) {
        const int col0 = wave << 4;
        const int nl = lane & 15;           // A: row M / B,C,D: col N
        const int g  = lane >> 4;           // lane half
        const int kb = g << 1;              // K sub-offset for this half

        v8f acc = {0.f,0.f,0.f,0.f,0.f,0.f,0.f,0.f};
        for (int k = 0; k < K; k += 4) {
            v2f a, b;
            // A 16x4 frag: lanes 0-15 K=0,1 ; lanes 16-31 K=2,3
            a.x = As[nl * lds + k + kb];
            a.y = As[nl * lds + k + kb + 1];
            // B 4x16 frag: VGPR v, lane-half g -> row v + 2g
            const float* Bp = B + (size_t)(k + kb) * ncols + col0 + nl;
            b.x = Bp[0];
            b.y = Bp[ncols];
            acc = __builtin_amdgcn_wmma_f32_16x16x4_f32(
                false, a, false, b, (short)0, acc, false, false);
        }
        const float bv = bias ? bias[col0 + nl] : 0.f;
        float* Cp = C + (size_t)row0 * ncols + col0 + nl;
#pragma unroll
        for (int v = 0; v < 8; ++v) {
            int mm = v + (g << 3);          // C/D: VGPR v, half g -> row v + 8g
            Cp[(size_t)mm * ncols] = acc[v] + bv;
        }
    }
}

// ---------------------------------------------------------------------------
// Ordered-int encoding of float for atomic max (monotonic in float order)
// ---------------------------------------------------------------------------
__device__ __forceinline__ unsigned encf(float f) {
    unsigned u = __float_as_uint(f);
    return (u & 0x80000000u) ? ~u : (u | 0x80000000u);
}
__device__ __forceinline__ float decf(unsigned k) {
    unsigned u = (k & 0x80000000u) ? (k & 0x7FFFFFFFu) : ~k;
    return __uint_as_float(u);
}

// el/er per (node, head) + init emax key (-inf) and denom
__global__ __launch_bounds__(256) void elr_init_kernel(
    const float* __restrict__ feat, const float* __restrict__ al,
    const float* __restrict__ ar, float* __restrict__ el, float* __restrict__ er,
    unsigned* __restrict__ emaxk, float* __restrict__ denom)
{
    int i = blockIdx.x * 256 + threadIdx.x;
    if (i >= N_NODES * H_HEADS) return;
    int h = i & (H_HEADS - 1);
    int node = i >> 2;
    const float* f = feat + (size_t)node * HID + h * D_DIM;
    const float* a = al + h * D_DIM;
    const float* b = ar + h * D_DIM;
    float sl = 0.f, sr = 0.f;
#pragma unroll
    for (int d = 0; d < D_DIM; ++d) { float v = f[d]; sl += v * a[d]; sr += v * b[d]; }
    el[i] = sl;
    er[i] = sr;
    emaxk[i] = 0x007FFFFFu;     // encf(-inf)
    denom[i] = 0.f;
}

__global__ __launch_bounds__(256) void edge_max_kernel(
    const int* __restrict__ src, const int* __restrict__ dst,
    const float* __restrict__ el, const float* __restrict__ er,
    unsigned* __restrict__ emaxk)
{
    int i = blockIdx.x * 256 + threadIdx.x;
    if (i >= E_EDGES * H_HEADS) return;
    int h = i & (H_HEADS - 1);
    int eidx = i >> 2;
    int s = src[eidx], d = dst[eidx];
    float e = el[s * H_HEADS + h] + er[d * H_HEADS + h];
    e = (e > 0.f) ? e : SLOPE * e;
    atomicMax(&emaxk[d * H_HEADS + h], encf(e));
}

__global__ __launch_bounds__(256) void edge_sum_kernel(
    const int* __restrict__ src, const int* __restrict__ dst,
    const float* __restrict__ el, const float* __restrict__ er,
    const unsigned* __restrict__ emaxk, float* __restrict__ denom,
    float* __restrict__ ee)
{
    int i = blockIdx.x * 256 + threadIdx.x;
    if (i >= E_EDGES * H_HEADS) return;
    int h = i & (H_HEADS - 1);
    int eidx = i >> 2;
    int s = src[eidx], d = dst[eidx];
    float e = el[s * H_HEADS + h] + er[d * H_HEADS + h];
    e = (e > 0.f) ? e : SLOPE * e;
    float x = expf(e - decf(emaxk[d * H_HEADS + h]));
    ee[i] = x;
    atomicAdd(&denom[d * H_HEADS + h], x);
}

__global__ __launch_bounds__(256) void edge_scatter_kernel(
    const int* __restrict__ src, const int* __restrict__ dst,
    const float* __restrict__ feat, const float* __restrict__ ee,
    const float* __restrict__ denom, float* __restrict__ out)
{
    int i = blockIdx.x * 256 + threadIdx.x;
    if (i >= E_EDGES * HID) return;
    int c = i & (HID - 1);
    int eidx = i >> 7;
    int h = c >> 5;                 // c / D_DIM
    int s = src[eidx], d = dst[eidx];
    float alpha = ee[eidx * H_HEADS + h] / fmaxf(denom[d * H_HEADS + h], 1e-9f);
    atomicAdd(&out[(size_t)d * HID + c], feat[(size_t)s * HID + c] * alpha);
}

// out[n][c] = sum_r b[r][c]  (accumulation target init, biases summed over R)
__global__ __launch_bounds__(256) void init_out_kernel(
    const float* __restrict__ b, float* __restrict__ out)
{
    int i = blockIdx.x * 256 + threadIdx.x;
    if (i >= N_NODES * HID) return;
    int c = i & (HID - 1);
    out[i] = b[c] + b[HID + c] + b[2 * HID + c] + b[3 * HID + c];
}

__global__ __launch_bounds__(256) void relu_kernel(float* __restrict__ x, int n)
{
    int i = blockIdx.x * 256 + threadIdx.x;
    if (i < n) x[i] = fmaxf(x[i], 0.f);
}

// ---------------------------------------------------------------------------
static void run_layer(const float* h_in, int F_in, const float* W,
                      const float* al, const float* ar, const float* b,
                      const int* src, const int* dst,
                      float* feat, float* el, float* er, unsigned* emaxk,
                      float* denom, float* ee, float* out, bool do_relu,
                      hipStream_t stream)
{
    init_out_kernel<<<(N_NODES * HID + 255) / 256, 256, 0, stream>>>(b, out);
    for (int r = 0; r < R_REL; ++r) {
        gemm_wmma_f32<<<N_NODES / 16, 256, 0, stream>>>(
            h_in, W + (size_t)r * F_in * HID, nullptr, feat, F_in, HID);
        elr_init_kernel<<<(N_NODES * H_HEADS + 255) / 256, 256, 0, stream>>>(
            feat, al + r * H_HEADS * D_DIM, ar + r * H_HEADS * D_DIM,
            el, er, emaxk, denom);
        const int* s = src + (size_t)r * E_EDGES;
        const int* d = dst + (size_t)r * E_EDGES;
        int eg = (E_EDGES * H_HEADS + 255) / 256;
        edge_max_kernel<<<eg, 256, 0, stream>>>(s, d, el, er, emaxk);
        edge_sum_kernel<<<eg, 256, 0, stream>>>(s, d, el, er, emaxk, denom, ee);
        edge_scatter_kernel<<<(E_EDGES * HID + 255) / 256, 256, 0, stream>>>(
            s, d, feat, ee, denom, out);
    }
    if (do_relu)
        relu_kernel<<<(N_NODES * HID + 255) / 256, 256, 0, stream>>>(out, N_NODES * HID);
}

extern "C" void kernel_launch(void* const* d_in, const int* in_sizes, int n_in,
                              void* d_out, int out_size, void* d_ws, size_t ws_size,
                              hipStream_t stream)
{
    (void)in_sizes; (void)n_in; (void)out_size; (void)ws_size;

    const float* x   = (const float*)d_in[0];
    const int*   src = (const int*)  d_in[1];
    const int*   dst = (const int*)  d_in[2];
    const float* W0  = (const float*)d_in[3];
    const float* al0 = (const float*)d_in[4];
    const float* ar0 = (const float*)d_in[5];
    const float* b0  = (const float*)d_in[6];
    const float* W1  = (const float*)d_in[7];
    const float* al1 = (const float*)d_in[8];
    const float* ar1 = (const float*)d_in[9];
    const float* b1  = (const float*)d_in[10];
    const float* W2  = (const float*)d_in[11];
    const float* al2 = (const float*)d_in[12];
    const float* ar2 = (const float*)d_in[13];
    const float* b2  = (const float*)d_in[14];
    const float* fcW = (const float*)d_in[15];
    const float* fcb = (const float*)d_in[16];
    float* out = (float*)d_out;

    char* ws = (char*)d_ws;
    size_t off = 0;
    auto take = [&](size_t bytes) -> void* {
        void* p = ws + off;
        off = (off + bytes + 255) & ~(size_t)255;
        return p;
    };
    float*    feat  = (float*)   take(sizeof(float)    * N_NODES * HID);
    float*    el    = (float*)   take(sizeof(float)    * N_NODES * H_HEADS);
    float*    er    = (float*)   take(sizeof(float)    * N_NODES * H_HEADS);
    unsigned* emaxk = (unsigned*)take(sizeof(unsigned) * N_NODES * H_HEADS);
    float*    denom = (float*)   take(sizeof(float)    * N_NODES * H_HEADS);
    float*    ee    = (float*)   take(sizeof(float)    * E_EDGES * H_HEADS);
    float*    h1    = (float*)   take(sizeof(float)    * N_NODES * HID);
    float*    h2    = (float*)   take(sizeof(float)    * N_NODES * HID);

    // layer 0: x (N x 256) -> h1, relu
    run_layer(x,  F_IN, W0, al0, ar0, b0, src, dst,
              feat, el, er, emaxk, denom, ee, h1, true, stream);
    // layer 1: h1 -> h2, relu
    run_layer(h1, HID,  W1, al1, ar1, b1, src, dst,
              feat, el, er, emaxk, denom, ee, h2, true, stream);
    // layer 2: h2 -> h1, no relu
    run_layer(h2, HID,  W2, al2, ar2, b2, src, dst,
              feat, el, er, emaxk, denom, ee, h1, false, stream);
    // final FC: (N x 128) @ (128 x 64) + fcb -> d_out
    gemm_wmma_f32<<<N_NODES / 16, 256, 0, stream>>>(h1, fcW, fcb, out, HID, F_OUT);
}